// cmdnet_36936718745670
// MI455X (gfx1250) — compile-verified
//
#include <hip/hip_runtime.h>
#include <math.h>

// CMDNet detector forward on gfx1250.
// B=16384 batches, NR=64, NT=32, M=4, 64 iterations.
// One wave (32 lanes) per batch. Gram matrix HH = Ht^T Ht and yH = Ht^T y are
// computed with chained V_WMMA_F32_16X16X4_F32 (full fp32 matrix cores).
// Iterations are VALU: lane j owns column j of HH (symmetric) in 32 VGPRs,
// xt is broadcast through LDS (ds_load_b128 broadcasts).

#define NUM_ITER 64
#define NRr 64
#define NTt 32
#define MM  4
#define WPB 8            // waves per block (8 batches / block)
#define LSTRIDE 33       // 32 + 1 padding -> conflict-free column reads

typedef __attribute__((ext_vector_type(2))) float v2f;
typedef __attribute__((ext_vector_type(8))) float v8f;

static __device__ __forceinline__ v8f wmma4(v2f a, v2f b, v8f c) {
  // D = A(16x4,f32) * B(4x16,f32) + C(16x16,f32)
  return __builtin_amdgcn_wmma_f32_16x16x4_f32(
      /*neg_a=*/false, a, /*neg_b=*/false, b,
      /*c_mod=*/(short)0, c, /*reuse_a=*/false, /*reuse_b=*/false);
}

__global__ __launch_bounds__(WPB * 32, 1)
void cmdnet_kernel(const float* __restrict__ yt,
                   const float* __restrict__ Ht,
                   const float* __restrict__ sigmat0,
                   const float* __restrict__ taui,
                   const float* __restrict__ delta,
                   const float* __restrict__ msym,
                   const float* __restrict__ alpha,
                   float* __restrict__ d_ft,
                   float* __restrict__ d_xt,
                   int Bn)
{
  __shared__ __align__(16) float s_hh[WPB][NTt * LSTRIDE];
  __shared__ float s_tau[NUM_ITER + 1];
  __shared__ float s_del[NUM_ITER];

  const int tid = threadIdx.x;
  if (tid < NUM_ITER + 1) s_tau[tid] = taui[tid];
  if (tid < NUM_ITER)     s_del[tid] = delta[tid];
  __syncthreads();                       // before any wave can exit (barrier safety)

  const int wave = tid >> 5;
  const int lane = tid & 31;
  const int half = lane >> 4;            // 0: lanes 0-15, 1: lanes 16-31
  const int col  = lane & 15;
  const int b    = blockIdx.x * WPB + wave;
  if (b >= Bn) return;

  const float* __restrict__ Hb = Ht + (size_t)b * (NRr * NTt);
  const float* __restrict__ yb = yt + (size_t)b * NRr;

  // ---------------- Gram + matched filter via fp32 WMMA ----------------
  // A-fragment layout (f32 16x4): lane L, vgpr v  ->  A[L%16, v + 2*(L/16)]
  // B assumed mirrored:           lane L, vgpr v  ->  B[v + 2*(L/16), L%16]
  v8f acc00 = {0,0,0,0,0,0,0,0};         // HH[ 0:16, 0:16]
  v8f acc01 = {0,0,0,0,0,0,0,0};         // HH[ 0:16,16:32]
  v8f acc10 = {0,0,0,0,0,0,0,0};         // HH[16:32, 0:16]
  v8f acc11 = {0,0,0,0,0,0,0,0};         // HH[16:32,16:32]
  v8f accY0 = {0,0,0,0,0,0,0,0};         // yH[ 0:16] replicated over rows
  v8f accY1 = {0,0,0,0,0,0,0,0};         // yH[16:32] replicated over rows

#pragma unroll
  for (int c = 0; c < NRr / 4; ++c) {
    const int r0 = 4 * c + 2 * half;     // this lane's pair of reduction rows
    const float* p = Hb + (size_t)r0 * NTt;
    v2f f0, f1, ay;
    f0[0] = __builtin_nontemporal_load(p + col);             // Ht[r0,   col]
    f0[1] = __builtin_nontemporal_load(p + NTt + col);       // Ht[r0+1, col]
    f1[0] = __builtin_nontemporal_load(p + 16 + col);        // Ht[r0,   16+col]
    f1[1] = __builtin_nontemporal_load(p + NTt + 16 + col);  // Ht[r0+1, 16+col]
    ay[0] = yb[r0];
    ay[1] = yb[r0 + 1];
    acc00 = wmma4(f0, f0, acc00);
    acc01 = wmma4(f0, f1, acc01);
    acc10 = wmma4(f1, f0, acc10);
    acc11 = wmma4(f1, f1, acc11);
    accY0 = wmma4(ay, f0, accY0);
    accY1 = wmma4(ay, f1, accY1);
  }

  // D-layout: lane L, vgpr v -> D[v + 8*(L/16), L%16].  Scatter HH to LDS.
  float* __restrict__ hh = s_hh[wave];
#pragma unroll
  for (int v = 0; v < 8; ++v) {
    const int r = v + 8 * half;
    hh[r * LSTRIDE + col]              = acc00[v];
    hh[r * LSTRIDE + 16 + col]         = acc01[v];
    hh[(16 + r) * LSTRIDE + col]       = acc10[v];
    hh[(16 + r) * LSTRIDE + 16 + col]  = acc11[v];
  }
  // yH[j]: every row of accY is identical; row held by this lane works.
  const float yH = (half == 0) ? accY0[0] : accY1[0];

  // Lane j pulls column j of HH (== row j, HH symmetric) into registers.
  // Same-wave DS ops are in-order, so stores above are visible.
  float HHcol[NTt];
#pragma unroll
  for (int i = 0; i < NTt; ++i)
    HHcol[i] = hh[i * LSTRIDE + lane];

  // ---------------- constants ----------------
  float mk[MM], la[MM];
#pragma unroll
  for (int k = 0; k < MM; ++k) { mk[k] = msym[k]; la[k] = __logf(alpha[k]); }
  const float s0   = sigmat0[b];
  const float sig2 = s0 * s0;

  // per-wave xt broadcast buffer (reuse of HH staging line; HH now in VGPRs)
  float* __restrict__ xbuf = hh;

  // ---------------- 64 gradient iterations ----------------
  float G[MM] = {0.f, 0.f, 0.f, 0.f};
  float ftv[MM];
  float xtv = 0.f;

#pragma unroll 1
  for (int t = 0; t < NUM_ITER; ++t) {
    const float ta = fabsf(s_tau[t]);
    const float sc = (t == 0) ? 1.0f : ta;

    // softmax over M=4 (max-subtracted)
    float z[MM];
#pragma unroll
    for (int k = 0; k < MM; ++k) z[k] = (la[k] + G[k]) * sc;
    const float zm = fmaxf(fmaxf(z[0], z[1]), fmaxf(z[2], z[3]));
    float fsum = 0.f;
#pragma unroll
    for (int k = 0; k < MM; ++k) { ftv[k] = __expf(z[k] - zm); fsum += ftv[k]; }
    const float inv = __builtin_amdgcn_rcpf(fsum);
    xtv = 0.f;
#pragma unroll
    for (int k = 0; k < MM; ++k) { ftv[k] *= inv; xtv += ftv[k] * mk[k]; }

    // broadcast xt across the wave via LDS, mat-vec against HH column
    xbuf[lane] = xtv;
    float xHH = 0.f;
    const float4* __restrict__ xq = (const float4*)xbuf;
#pragma unroll
    for (int q = 0; q < NTt / 4; ++q) {
      const float4 xv = xq[q];                 // ds_load_b128 broadcast
      xHH = fmaf(xv.x, HHcol[4 * q + 0], xHH);
      xHH = fmaf(xv.y, HHcol[4 * q + 1], xHH);
      xHH = fmaf(xv.z, HHcol[4 * q + 2], xHH);
      xHH = fmaf(xv.w, HHcol[4 * q + 3], xHH);
    }
    const float e = xHH - yH;
    const float d = s_del[t];
#pragma unroll
    for (int k = 0; k < MM; ++k) {
      const float gl = sig2 * (1.0f - __expf(-G[k]))
                     + ta * ftv[k] * (mk[k] - xtv) * e;
      G[k] -= d * gl;
    }
  }

  // ---------------- final layer: softmax + soft symbol only ----------------
  {
    const float ta = fabsf(s_tau[NUM_ITER]);
    float z[MM];
#pragma unroll
    for (int k = 0; k < MM; ++k) z[k] = (la[k] + G[k]) * ta;
    const float zm = fmaxf(fmaxf(z[0], z[1]), fmaxf(z[2], z[3]));
    float fsum = 0.f;
#pragma unroll
    for (int k = 0; k < MM; ++k) { ftv[k] = __expf(z[k] - zm); fsum += ftv[k]; }
    const float inv = __builtin_amdgcn_rcpf(fsum);
    xtv = 0.f;
#pragma unroll
    for (int k = 0; k < MM; ++k) { ftv[k] *= inv; xtv += ftv[k] * mk[k]; }
  }

  // outputs: ft [B,NT,M] then xt [B,NT], concatenated flat
  float4 fo;
  fo.x = ftv[0]; fo.y = ftv[1]; fo.z = ftv[2]; fo.w = ftv[3];
  *(float4*)(d_ft + (size_t)b * (NTt * MM) + (size_t)lane * MM) = fo;
  d_xt[(size_t)b * NTt + lane] = xtv;
}

extern "C" void kernel_launch(void* const* d_in, const int* in_sizes, int n_in,
                              void* d_out, int out_size, void* d_ws, size_t ws_size,
                              hipStream_t stream) {
  const float* yt = (const float*)d_in[0];
  const float* Ht = (const float*)d_in[1];
  const float* sg = (const float*)d_in[2];
  const float* ta = (const float*)d_in[3];
  const float* de = (const float*)d_in[4];
  const float* m  = (const float*)d_in[5];
  const float* al = (const float*)d_in[6];
  const int Bn = in_sizes[2];                  // sigmat0 length == batch count

  float* ftp = (float*)d_out;                  // [B, NT, M]
  float* xtp = ftp + (size_t)Bn * NTt * MM;    // [B, NT]

  dim3 grid((Bn + WPB - 1) / WPB), block(WPB * 32);
  hipLaunchKernelGGL(cmdnet_kernel, grid, block, 0, stream,
                     yt, Ht, sg, ta, de, m, al, ftp, xtp, Bn);
}